// GraphSIR_30597347016802
// MI455X (gfx1250) — compile-verified
//
#include <hip/hip_runtime.h>

// GraphSIR derivative on MI455X (gfx1250).
// Memory-bound scatter-add: ~66 MB streamed/call -> ~2.8us floor @ 23.3 TB/s HBM;
// whole working set (~68 MB) fits the 192 MB L2, so graph replays run at L2 BW.
// WMMA is inapplicable (irregular int64-indexed scatter, ~0.1 FLOP/byte); the
// gfx1250 paths exercised: global_load_b128 (edge pairs), no-return
// global_atomic_add_f32 (L2 atomic units, STOREcnt), global_prefetch_b8.

// Pass 1: per-node terms; fully overwrites d_out every call; compacts i = s[:,1]
// into a dense 400 KB array in d_ws (3x denser gather cachelines for pass 2).
__global__ __launch_bounds__(256) void sir_node_kernel(
    const float* __restrict__ s,
    const float* __restrict__ intra_b,
    const float* __restrict__ intra_k,
    float* __restrict__ out,
    float* __restrict__ i_copy,   // may be null if ws too small
    int n)
{
    int idx = blockIdx.x * blockDim.x + threadIdx.x;
    if (idx >= n) return;
    float s0 = s[3 * idx + 0];
    float i  = s[3 * idx + 1];
    float si = s0 * i * intra_b[idx];
    float ri = i * intra_k[idx];
    out[3 * idx + 0] = -si;
    out[3 * idx + 1] = si - ri;
    out[3 * idx + 2] = ri;
    if (i_copy) i_copy[idx] = i;
}

// Pass 2: per-edge diffusion, scatter-add into out[:,1].
// IST is a compile-time gather stride: iv[IST*k] == i[k]
// (IST==1: compacted ws copy; IST==3: read s[:,1] in place).
template <int IST>
__global__ __launch_bounds__(256) void sir_edge_kernel(
    const float* __restrict__ iv,
    const float* __restrict__ inter_b,
    const longlong2* __restrict__ adj,
    float* __restrict__ out,
    int e_count)
{
    int tot = gridDim.x * blockDim.x;
    for (int e = blockIdx.x * blockDim.x + threadIdx.x; e < e_count; e += tot) {
        // Stream next tile ahead (speculative; silently dropped past the end)
        // -> global_prefetch_b8.
        __builtin_prefetch(&adj[e + tot], 0, 3);
        __builtin_prefetch(&inter_b[e + tot], 0, 3);

        longlong2 p = adj[e];            // one global_load_b128 per edge
        int a = (int)p.x;
        int b = (int)p.y;
        float w  = inter_b[e];
        float ia = iv[IST * a];          // b32 gathers, L2/WGP$-resident target
        float ib = iv[IST * b];
        float d  = w * (ib - ia);
        // Unused return -> no-return global_atomic_add_f32.
        atomicAdd(&out[3 * a + 1],  d);
        atomicAdd(&out[3 * b + 1], -d);
    }
}

extern "C" void kernel_launch(void* const* d_in, const int* in_sizes, int n_in,
                              void* d_out, int out_size, void* d_ws, size_t ws_size,
                              hipStream_t stream) {
    // inputs: 0=t, 1=s[N,3] f32, 2=intra_b[N] f32, 3=intra_k[N] f32,
    //         4=inter_b[E] f32, 5=inter_adj[E,2] int64
    const float*     s       = (const float*)d_in[1];
    const float*     intra_b = (const float*)d_in[2];
    const float*     intra_k = (const float*)d_in[3];
    const float*     inter_b = (const float*)d_in[4];
    const longlong2* adj     = (const longlong2*)d_in[5];
    float* out = (float*)d_out;

    const int n = in_sizes[2];
    const int e = in_sizes[4];

    const bool use_ws = (ws_size >= (size_t)n * sizeof(float));
    float* i_copy = use_ws ? (float*)d_ws : nullptr;

    // Pass 1: node terms + i compaction (rewrites all of d_out / i_copy each call,
    // so the harness's 0xAA poison is always overwritten before pass 2 reads it).
    sir_node_kernel<<<(n + 255) / 256, 256, 0, stream>>>(
        s, intra_b, intra_k, out, i_copy, n);

    // Pass 2: edge scatter. 4096 blocks x 256 thr (8 wave32/block) = 1M threads,
    // ~3 grid-stride iterations each -> prefetch one tile ahead; saturates HBM/L2.
    const int blocks = 4096;
    if (use_ws) {
        sir_edge_kernel<1><<<blocks, 256, 0, stream>>>(i_copy, inter_b, adj, out, e);
    } else {
        sir_edge_kernel<3><<<blocks, 256, 0, stream>>>(s + 1, inter_b, adj, out, e);
    }
}